// GATLayer_61838939127953
// MI455X (gfx1250) — compile-verified
//
#include <hip/hip_runtime.h>
#include <hip/hip_bf16.h>

// ---------------------------------------------------------------------------
// GAT layer for MI455X (gfx1250, wave32).
//   N=100000 nodes, E=1600000 edges, IN=128, H=4 heads, F=16 feats/head.
// Memory-bound workload; node-transform GEMM done with V_WMMA_F32_16X16X4_F32
// (full f32 precision, 32 chained WMMAs per 16x16 tile over K=128).
// ---------------------------------------------------------------------------

#define N_NODES   100000
#define E_EDGES   1600000
#define IN_F      128
#define H_HEADS   4
#define F_OUT     16
#define HF        64            // H*F
#define NEG_SLOPE 0.01f

typedef float v2f __attribute__((ext_vector_type(2)));
typedef float v8f __attribute__((ext_vector_type(8)));

// Order-preserving float -> uint encoding so we can use integer atomicMax.
__device__ __forceinline__ unsigned fenc(float f) {
    unsigned u = __float_as_uint(f);
    return (u & 0x80000000u) ? ~u : (u | 0x80000000u);
}
__device__ __forceinline__ float fdec(unsigned k) {
    return (k & 0x80000000u) ? __uint_as_float(k ^ 0x80000000u)
                             : __uint_as_float(~k);
}

// ---------------------------------------------------------------------------
// 0) Zero out / denom, reset global max (harness poisons buffers with 0xAA).
// ---------------------------------------------------------------------------
__global__ void gat_init(float* __restrict__ out, float* __restrict__ denom,
                         unsigned* __restrict__ maxenc) {
    size_t i = (size_t)blockIdx.x * blockDim.x + threadIdx.x;
    if (i < (size_t)N_NODES * HF)      out[i]   = 0.0f;
    if (i < (size_t)N_NODES * H_HEADS) denom[i] = 0.0f;
    if (i == 0) *maxenc = 0u;   // any finite float encodes > 0
}

// ---------------------------------------------------------------------------
// 1) nt = x @ W^T   (N x 128) @ (128 x 64) via V_WMMA_F32_16X16X4_F32.
//    Block = 128 threads = 4 waves; wave w computes the 16x16 tile at
//    rows [blockIdx.x*16, +16), cols [w*16, +16).  N % 16 == 0 -> EXEC all 1s.
//
//    f32 A 16x4 layout (ISA 7.12.2): lane L holds M = L&15,
//      VGPR v holds K = 2*(L>>4) + v.  B (4x16) symmetric with N = L&15.
//    C/D: lane L, VGPR r -> M = r + 8*(L>>4), N = L&15.
// ---------------------------------------------------------------------------
__global__ __launch_bounds__(128)
void gat_transform_wmma(const float* __restrict__ x,
                        const float* __restrict__ W,   // (64,128) row-major
                        float* __restrict__ nt) {
    const int lane = threadIdx.x & 31;
    const int wave = threadIdx.x >> 5;
    const int m    = lane & 15;
    const int kk   = lane >> 4;                 // 0 or 1
    const int row  = blockIdx.x * 16 + m;       // exact: 6250*16 == N
    const int ncol = wave * 16 + m;             // B: N index == lane&15

    const float* xrow = x + (size_t)row  * IN_F + 2 * kk;
    const float* wrow = W + (size_t)ncol * IN_F + 2 * kk;

    v8f c = {};
#pragma unroll
    for (int k0 = 0; k0 < IN_F; k0 += 4) {
        v2f a = *(const v2f*)(xrow + k0);       // K = k0 + 2*kk + {0,1}
        v2f b = *(const v2f*)(wrow + k0);
        c = __builtin_amdgcn_wmma_f32_16x16x4_f32(
                /*neg_a=*/false, a, /*neg_b=*/false, b,
                /*c_mod=*/(short)0, c, /*reuse_a=*/false, /*reuse_b=*/false);
    }

    const int half = lane >> 4;
#pragma unroll
    for (int r = 0; r < 8; ++r) {
        int mm = r + 8 * half;
        nt[(size_t)(blockIdx.x * 16 + mm) * HF + wave * 16 + (lane & 15)] = c[r];
    }
}

// ---------------------------------------------------------------------------
// 2) Per-edge attention logits + global max.
//    One wave per edge. pairs (len 128) is head-major: for head block h':
//      pairs[h'*32 + f]      = s[h'*16 + f]   (f < 16)
//      pairs[h'*32 + 16 + f] = t[h'*16 + f]
//    Lane L covers pairs[4L .. 4L+3] (always within one s/t half) as a float4.
// ---------------------------------------------------------------------------
__global__ __launch_bounds__(256)
void gat_edge_scores(const float* __restrict__ nt,
                     const float* __restrict__ a_w,      // (4,128) row-major
                     const int*   __restrict__ src,
                     const int*   __restrict__ tgt,
                     float* __restrict__ aw,             // (E,4)
                     unsigned* __restrict__ maxenc) {
    __shared__ unsigned smax;
    if (threadIdx.x == 0) smax = 0u;
    __syncthreads();

    const int lane = threadIdx.x & 31;
    const int wv   = threadIdx.x >> 5;
    const int e    = blockIdx.x * 8 + wv;

    float res[H_HEADS] = {0.f, 0.f, 0.f, 0.f};
    if (e < E_EDGES) {
        const int vs = src[e];
        const int vt = tgt[e];
        const int j  = lane * 4;          // pair index base
        const int hp = j >> 5;            // which head block
        const int off = j & 31;
        const float* base = (off < 16)
            ? (nt + (size_t)vs * HF + hp * F_OUT + off)
            : (nt + (size_t)vt * HF + hp * F_OUT + (off - 16));
        const float4 v = *(const float4*)base;
#pragma unroll
        for (int h = 0; h < H_HEADS; ++h) {
            const float* ar = a_w + h * (2 * HF) + j;
            res[h] = fmaf(v.x, ar[0], fmaf(v.y, ar[1],
                     fmaf(v.z, ar[2], v.w * ar[3])));
        }
    }
    // wave32 butterfly reduction (all lanes end with the full sums)
#pragma unroll
    for (int d = 16; d >= 1; d >>= 1) {
#pragma unroll
        for (int h = 0; h < H_HEADS; ++h)
            res[h] += __shfl_xor(res[h], d, 32);
    }
    if (e < E_EDGES && lane == 0) {
        float mx = res[0];
#pragma unroll
        for (int h = 0; h < H_HEADS; ++h) {
            aw[(size_t)e * H_HEADS + h] = res[h];
            mx = fmaxf(mx, res[h]);
        }
        atomicMax(&smax, fenc(mx));
    }
    __syncthreads();
    if (threadIdx.x == 0) atomicMax(maxenc, smax);
}

// ---------------------------------------------------------------------------
// 3) ex = exp(leaky_relu(aw - gmax)); denom[tgt,h] += ex.  In-place on aw.
// ---------------------------------------------------------------------------
__global__ __launch_bounds__(256)
void gat_softmax_denom(float* __restrict__ awex,        // (E,4), in-place
                       const int* __restrict__ tgt,
                       float* __restrict__ denom,       // (N,4)
                       const unsigned* __restrict__ maxenc) {
    const int e = blockIdx.x * blockDim.x + threadIdx.x;
    if (e >= E_EDGES) return;
    const float gmax = fdec(*maxenc);
    const int vt = tgt[e];
    float4 a = *(const float4*)(awex + (size_t)e * H_HEADS);
    float v[H_HEADS] = {a.x, a.y, a.z, a.w};
#pragma unroll
    for (int h = 0; h < H_HEADS; ++h) {
        float z = v[h] - gmax;
        z = (z > 0.0f) ? z : NEG_SLOPE * z;
        float ev = __expf(z);
        awex[(size_t)e * H_HEADS + h] = ev;
        atomicAdd(&denom[(size_t)vt * H_HEADS + h], ev);
    }
}

// ---------------------------------------------------------------------------
// 4) out[tgt] += alpha * s.  One wave per edge; each lane handles 2 of the
//    64 (h,f) outputs.  All atomics land in L2 (out is 25.6 MB, L2-resident).
// ---------------------------------------------------------------------------
__global__ __launch_bounds__(256)
void gat_aggregate(const float* __restrict__ nt,
                   const float* __restrict__ ex,        // (E,4)
                   const float* __restrict__ denom,     // (N,4)
                   const int*   __restrict__ src,
                   const int*   __restrict__ tgt,
                   float* __restrict__ out) {
    const int lane = threadIdx.x & 31;
    const int wv   = threadIdx.x >> 5;
    const int e    = blockIdx.x * 8 + wv;
    if (e >= E_EDGES) return;
    const int vs = src[e];
    const int vt = tgt[e];
#pragma unroll
    for (int c = 0; c < 2; ++c) {
        const int idx = lane + 32 * c;   // 0..63 over (h,f)
        const int h   = idx >> 4;
        const float alpha = ex[(size_t)e * H_HEADS + h] /
                            (denom[(size_t)vt * H_HEADS + h] + 1e-8f);
        const float sv = nt[(size_t)vs * HF + idx];
        atomicAdd(&out[(size_t)vt * HF + idx], alpha * sv);
    }
}

// ---------------------------------------------------------------------------
extern "C" void kernel_launch(void* const* d_in, const int* in_sizes, int n_in,
                              void* d_out, int out_size, void* d_ws, size_t ws_size,
                              hipStream_t stream) {
    const float* x   = (const float*)d_in[0];   // (N,128)
    const float* W_w = (const float*)d_in[1];   // (64,128)
    const float* a_w = (const float*)d_in[2];   // (4,128)
    const int*   ei  = (const int*)d_in[3];     // (2,E)
    const int* src = ei;
    const int* tgt = ei + E_EDGES;
    float* out = (float*)d_out;                 // (N,64)

    // Workspace layout (floats): nt | aw/ex | denom | maxenc
    float* nt    = (float*)d_ws;
    float* awex  = nt   + (size_t)N_NODES * HF;
    float* denom = awex + (size_t)E_EDGES * H_HEADS;
    unsigned* maxenc = (unsigned*)(denom + (size_t)N_NODES * H_HEADS);

    (void)in_sizes; (void)n_in; (void)out_size; (void)ws_size;

    {   // 0) init
        size_t total = (size_t)N_NODES * HF;
        int blocks = (int)((total + 255) / 256);
        gat_init<<<blocks, 256, 0, stream>>>(out, denom, maxenc);
    }
    {   // 1) node transform via WMMA f32 16x16x4
        gat_transform_wmma<<<N_NODES / 16, 128, 0, stream>>>(x, W_w, nt);
    }
    {   // 2) edge logits + global max (8 edges per 256-thread block)
        gat_edge_scores<<<(E_EDGES + 7) / 8, 256, 0, stream>>>(
            nt, a_w, src, tgt, awex, maxenc);
    }
    {   // 3) exp + segment-sum denominator
        gat_softmax_denom<<<(E_EDGES + 255) / 256, 256, 0, stream>>>(
            awex, tgt, denom, maxenc);
    }
    {   // 4) weighted scatter aggregation
        gat_aggregate<<<(E_EDGES + 7) / 8, 256, 0, stream>>>(
            nt, awex, denom, src, tgt, out);
    }
}